// SparsemaxAllocator_59081570124230
// MI455X (gfx1250) — compile-verified
//
#include <hip/hip_runtime.h>
#include <stdint.h>

// Capped-simplex projection (sparsemax with per-asset cap) for MI455X / gfx1250.
//
//  * 16 rows per workgroup, 1024 threads (32 waves, wave32).
//  * Tile (16 x 4096 fp32, 256KB) staged into LDS by the Tensor Data Mover
//    (one per-row descriptor, TENSORcnt-tracked), rows padded to 4100 dwords
//    so the LDS->VGPR fill is bank-conflict free.
//  * lane L owns row (L & 15); 64 elements/thread live in VGPRs for all 60
//    bisection iterations (no HBM/LDS traffic in the hot loop; hot loop is
//    ~2 VALU slots/element: v_pk_add_f32 + v_med3_num_f32).
//  * Per-iteration row sums: fp32 V_WMMA_F32_16X16X4_F32 with B = ones folds
//    the 32 lane-partials of each wave into 16 per-row partials on the matrix
//    pipe; a tiny LDS pass combines the 32 waves.
//  * Final tau recomputed analytically from the active set (KKT-consistent,
//    as the reference), outputs stored as coalesced float4.

typedef __attribute__((ext_vector_type(2))) float v2f;
typedef __attribute__((ext_vector_type(8))) float v8f;
typedef unsigned int u32x4 __attribute__((ext_vector_type(4)));
typedef int          i32x4 __attribute__((ext_vector_type(4)));
typedef int          i32x8 __attribute__((ext_vector_type(8)));

namespace {
constexpr int   kRowsPerWG = 16;
constexpr int   kAssets    = 4096;
constexpr int   kThreads   = 1024;                    // 32 waves of 32
constexpr int   kWaves     = kThreads / 32;           // 32
constexpr int   kElems     = (kRowsPerWG * kAssets) / kThreads;  // 64 / thread
constexpr int   kVec       = kElems / 4;              // 16 x float4 in VGPRs
constexpr int   kRowStrideF = kAssets + 4;            // 4100 dwords: bank-skewed
constexpr int   kTileF      = kRowsPerWG * kRowStrideF;
constexpr int   kRedF       = kWaves * kRowsPerWG;    // 512 floats per scratch arr
constexpr int   kSmemF      = kTileF + 3 * kRedF;
constexpr float kU         = 0.02f;
constexpr float kEps       = 1e-7f;
constexpr int   kIters     = 60;
}

// ---- TDM: DMA one tensor row (n_cols fp32, contiguous) into LDS ------------
// D# group0/group1 built per CDNA5 ISA:
//   g0: count=1 | lds_addr | global_addr[56:0] | type=2
//   g1: data_size=4B, tensor_dim0=n_cols, tensor_dim1=1, tile_dim0=n_cols,
//       tile_dim1=1, tensor_dim0_stride=n_cols, no multicast/pad/iterate.
__device__ __forceinline__ void tdm_load_row_to_lds(uint32_t lds_byte_off,
                                                    const float* gsrc,
                                                    uint32_t n_cols) {
  const uint64_t ga = (uint64_t)(uintptr_t)gsrc;
  u32x4 g0;
  g0[0] = 1u;                                         // count=1, user-mode D#
  g0[1] = lds_byte_off;                               // LDS destination
  g0[2] = (uint32_t)(ga & 0xFFFFFFFFu);               // global_addr[31:0]
  g0[3] = (uint32_t)((ga >> 32) & 0x01FFFFFFu)        // global_addr[56:32]
          | 0x80000000u;                              // type = 2 ("image")
  i32x8 g1;
  g1[0] = (int)(2u << 16);                            // data_size = 4 bytes
  g1[1] = (int)((n_cols & 0xFFFFu) << 16);            // tensor_dim0 lo16
  g1[2] = (int)((n_cols >> 16) | (1u << 16));         // dim0 hi16 | tensor_dim1=1
  g1[3] = (int)((n_cols & 0xFFFFu) << 16);            // tile_dim0
  g1[4] = 1;                                          // tile_dim1=1, tile_dim2=0
  g1[5] = (int)n_cols;                                // tensor_dim0_stride lo32
  g1[6] = 0;
  g1[7] = 0;
  const i32x4 z4 = {0, 0, 0, 0};
#if defined(__clang_major__) && (__clang_major__ >= 23)
  const i32x8 z8 = {0, 0, 0, 0, 0, 0, 0, 0};
  __builtin_amdgcn_tensor_load_to_lds(g0, g1, z4, z4, z8, 0);   // clang-23 lane
#else
  __builtin_amdgcn_tensor_load_to_lds(g0, g1, z4, z4, 0);       // ROCm 7.2 lane
#endif
}

// ---- Matrix-pipe cross-lane row reduction -----------------------------------
// A (16x4 f32, 2 VGPRs): lanes 0-15 supply K0/K1, lanes 16-31 K2/K3.
// a = (acc, 0), B = ones  =>  D[r][*] = acc(lane r) + acc(lane r+16).
// D layout: VGPR j = row j (lanes 0-15) / row j+8 (lanes 16-31).
__device__ __forceinline__ void wmma_row_reduce_store(float acc,
                                                      float* p_wave_base,
                                                      int lane) {
  v2f a; a[0] = acc;  a[1] = 0.0f;
  v2f b; b[0] = 1.0f; b[1] = 1.0f;
  v8f c = {0.0f, 0.0f, 0.0f, 0.0f, 0.0f, 0.0f, 0.0f, 0.0f};
  c = __builtin_amdgcn_wmma_f32_16x16x4_f32(false, a, false, b,
                                            (short)0, c, false, false);
  if ((lane & 15) == 0) {                 // lanes 0 / 16 hold rows 0-7 / 8-15
    const int rbase = (lane >> 4) << 3;
    #pragma unroll
    for (int j = 0; j < 8; ++j) p_wave_base[rbase + j] = c[j];
  }
}

__global__ void __launch_bounds__(kThreads)
capped_sparsemax_kernel(const float* __restrict__ x,
                        const float* __restrict__ temp,
                        float* __restrict__ out) {
  extern __shared__ float smem[];
  float* tile = smem;                     // 16 x 4100 fp32 (bank-skewed rows)
  float* P0   = smem + kTileF;            // 3 x 512 fp32 reduction scratch
  float* P1   = P0 + kRedF;
  float* P2   = P1 + kRedF;

  const int tid  = (int)threadIdx.x;
  const int wv   = tid >> 5;
  const int lane = tid & 31;
  const int r    = lane & 15;                   // row within the 16-row tile
  const int grp  = (wv << 1) | (lane >> 4);     // 64 column groups of 64 cols

  const size_t row  = (size_t)blockIdx.x * kRowsPerWG + (size_t)r;
  const size_t base = row * (size_t)kAssets + (size_t)grp * (size_t)kElems;

  // ---- TDM fill: 16 per-row descriptors issued by wave 0 -------------------
  if (tid == 0) {
    const uint32_t lds_base = (uint32_t)(uintptr_t)tile;  // flat LDS: low 32b
    const float* grow = x + (size_t)blockIdx.x * kRowsPerWG * kAssets;
    #pragma unroll
    for (int rr = 0; rr < kRowsPerWG; ++rr) {
      tdm_load_row_to_lds(lds_base + (uint32_t)rr * (kRowStrideF * 4u),
                          grow + (size_t)rr * kAssets, (uint32_t)kAssets);
    }
    __builtin_amdgcn_s_wait_tensorcnt(0);       // TENSORcnt == 0: tile resident
  }
  __syncthreads();

  const float t = temp[row];

  // ---- LDS -> VGPR fill (conflict-free: bank = 4r + 4i), z = x / t ---------
  const float4* __restrict__ lsrc =
      (const float4*)(tile + (size_t)r * kRowStrideF + (size_t)grp * kElems);
  float4 zv[kVec];
  float mn = 3.4028235e38f, mx = -3.4028235e38f;
  #pragma unroll
  for (int i = 0; i < kVec; ++i) {
    float4 v = lsrc[i];
    v.x /= t; v.y /= t; v.z /= t; v.w /= t;     // exact IEEE div (one-time)
    zv[i] = v;
    mn = fminf(mn, fminf(fminf(v.x, v.y), fminf(v.z, v.w)));
    mx = fmaxf(mx, fmaxf(fmaxf(v.x, v.y), fmaxf(v.z, v.w)));
  }

  // ---- Row min/max across the workgroup (one-time) -------------------------
  mn = fminf(mn, __shfl_xor(mn, 16));           // fold lane r+16 into lane r
  mx = fmaxf(mx, __shfl_xor(mx, 16));
  if (lane < 16) {
    P0[wv * 16 + lane] = mn;
    P1[wv * 16 + lane] = mx;
  }
  __syncthreads();
  float lo = 3.4028235e38f, hi = -3.4028235e38f;
  #pragma unroll
  for (int w2 = 0; w2 < kWaves; ++w2) {
    lo = fminf(lo, P0[w2 * 16 + r]);
    hi = fmaxf(hi, P1[w2 * 16 + r]);
  }
  lo -= 1.0f;                                   // reference: lo = min(z) - 1
  __syncthreads();

  // ---- Bisection: data stays in VGPRs, reduce via WMMA + LDS ---------------
  for (int it = 0; it < kIters; ++it) {
    const float mid = 0.5f * (lo + hi);
    float a0 = 0.0f, a1 = 0.0f, a2 = 0.0f, a3 = 0.0f;  // break add-latency chain
    #pragma unroll
    for (int i = 0; i < kVec; ++i) {
      const float4 v = zv[i];
      a0 += fminf(fmaxf(v.x - mid, 0.0f), kU);
      a1 += fminf(fmaxf(v.y - mid, 0.0f), kU);
      a2 += fminf(fmaxf(v.z - mid, 0.0f), kU);
      a3 += fminf(fmaxf(v.w - mid, 0.0f), kU);
    }
    wmma_row_reduce_store((a0 + a1) + (a2 + a3), P0 + wv * 16, lane);
    __syncthreads();
    float s = 0.0f;
    #pragma unroll
    for (int w2 = 0; w2 < kWaves; ++w2) s += P0[w2 * 16 + r];
    const bool big = (s > 1.0f);                // identical across a row's threads
    lo = big ? mid : lo;
    hi = big ? hi : mid;
    __syncthreads();
  }

  // ---- Analytic tau from the active set (matches reference KKT recompute) --
  const float tau0 = 0.5f * (lo + hi);
  float sf = 0.0f, nf = 0.0f, nc = 0.0f;
  #pragma unroll
  for (int i = 0; i < kVec; ++i) {
    const float4 v = zv[i];
    const float e[4] = {v.x, v.y, v.z, v.w};
    #pragma unroll
    for (int jj = 0; jj < 4; ++jj) {
      const float w0 = fminf(fmaxf(e[jj] - tau0, 0.0f), kU);
      const bool fr = (w0 > kEps) && (w0 < kU - kEps);
      const bool cp = (w0 >= kU - kEps);
      sf += fr ? e[jj] : 0.0f;
      nf += fr ? 1.0f : 0.0f;
      nc += cp ? 1.0f : 0.0f;
    }
  }
  wmma_row_reduce_store(sf, P0 + wv * 16, lane);
  wmma_row_reduce_store(nf, P1 + wv * 16, lane);
  wmma_row_reduce_store(nc, P2 + wv * 16, lane);
  __syncthreads();
  float Sf = 0.0f, Nf = 0.0f, Nc = 0.0f;
  #pragma unroll
  for (int w2 = 0; w2 < kWaves; ++w2) {
    Sf += P0[w2 * 16 + r];
    Nf += P1[w2 * 16 + r];
    Nc += P2[w2 * 16 + r];
  }
  const float tau = (Sf + kU * Nc - 1.0f) / fmaxf(Nf, 1.0f);

  // ---- Emit outputs straight from registers (coalesced float4) -------------
  float4* __restrict__ dst = (float4*)(out + base);
  #pragma unroll
  for (int i = 0; i < kVec; ++i) {
    const float4 v = zv[i];
    const float e[4] = {v.x, v.y, v.z, v.w};
    float o[4];
    #pragma unroll
    for (int jj = 0; jj < 4; ++jj) {
      const float w0 = fminf(fmaxf(e[jj] - tau0, 0.0f), kU);
      const bool fr = (w0 > kEps) && (w0 < kU - kEps);
      const bool cp = (w0 >= kU - kEps);
      o[jj] = fr ? (e[jj] - tau) : (cp ? kU : 0.0f);
    }
    float4 ov; ov.x = o[0]; ov.y = o[1]; ov.z = o[2]; ov.w = o[3];
    dst[i] = ov;
  }
}

extern "C" void kernel_launch(void* const* d_in, const int* in_sizes, int n_in,
                              void* d_out, int out_size, void* d_ws, size_t ws_size,
                              hipStream_t stream) {
  const float* x    = (const float*)d_in[0];   // (N_SAMPLES, N_ASSETS) fp32
  const float* temp = (const float*)d_in[1];   // (N_SAMPLES,) fp32
  float* out        = (float*)d_out;

  const int n_samples = in_sizes[1];                 // temperature count
  const int grid      = n_samples / kRowsPerWG;      // 16384/16 = 1024 WGs
  const size_t smem_bytes = (size_t)kSmemF * sizeof(float);  // ~268 KB < 320 KB

  hipLaunchKernelGGL(capped_sparsemax_kernel, dim3(grid), dim3(kThreads),
                     smem_bytes, stream, x, temp, out);
}